// SOC_Solver_19808389169330
// MI455X (gfx1250) — compile-verified
//
#include <hip/hip_runtime.h>
#include <hip/hip_bf16.h>
#include <math.h>

#define TN 201
#define NSTEPS 200
#define BATCH 4096
#define DIM 20
#define HIDDEN 128
#define BTILE 16
#define BLOCK 256

typedef float v2f __attribute__((ext_vector_type(2)));
typedef float v8f __attribute__((ext_vector_type(8)));

// Branch-free tanh: prefer gfx1250 hardware v_tanh_f32.
#if __has_builtin(__builtin_amdgcn_tanhf)
__device__ __forceinline__ float tanh_fast(float x) { return __builtin_amdgcn_tanhf(x); }
#elif __has_builtin(__builtin_amdgcn_tanh_f32)
__device__ __forceinline__ float tanh_fast(float x) { return __builtin_amdgcn_tanh_f32(x); }
#else
__device__ __forceinline__ float tanh_fast(float x) {
    // tanh(x) = 1 - 2/(exp(2x)+1); saturates correctly at +/-inf, branch-free.
    float e2x = __expf(2.0f * x);
    return 1.0f - 2.0f * __builtin_amdgcn_rcpf(e2x + 1.0f);
}
#endif

// ---------------------------------------------------------------------------
// Kernel 1: invert sigma (20x20, partial-pivot Gauss-Jordan) and precompute
//   M2[i][j] = -sqrt(LMBD) * sum_k A[i][k] * inv(sigma)[j][k]
//   M3[i][j] = -            sum_k A[i][k] * inv(sigma)[j][k]
// ---------------------------------------------------------------------------
__global__ void soc_precompute(const float* __restrict__ sigma,
                               const float* __restrict__ A,
                               float* __restrict__ M2,
                               float* __restrict__ M3) {
    if (threadIdx.x != 0 || blockIdx.x != 0) return;
    float m[DIM][DIM];
    float inv[DIM][DIM];
    for (int i = 0; i < DIM; ++i)
        for (int j = 0; j < DIM; ++j) {
            m[i][j] = sigma[i * DIM + j];
            inv[i][j] = (i == j) ? 1.0f : 0.0f;
        }
    for (int col = 0; col < DIM; ++col) {
        int p = col;
        float best = fabsf(m[col][col]);
        for (int r = col + 1; r < DIM; ++r) {
            float v = fabsf(m[r][col]);
            if (v > best) { best = v; p = r; }
        }
        if (p != col) {
            for (int j = 0; j < DIM; ++j) {
                float t = m[col][j]; m[col][j] = m[p][j]; m[p][j] = t;
                t = inv[col][j]; inv[col][j] = inv[p][j]; inv[p][j] = t;
            }
        }
        float ip = 1.0f / m[col][col];
        for (int j = 0; j < DIM; ++j) { m[col][j] *= ip; inv[col][j] *= ip; }
        for (int r = 0; r < DIM; ++r) {
            if (r == col) continue;
            float f = m[r][col];
            for (int j = 0; j < DIM; ++j) {
                m[r][j] -= f * m[col][j];
                inv[r][j] -= f * inv[col][j];
            }
        }
    }
    const float sqrtL = 1.0f; // sqrt(LMBD), LMBD = 1
    for (int i = 0; i < DIM; ++i)
        for (int j = 0; j < DIM; ++j) {
            float s = 0.0f;
            for (int k = 0; k < DIM; ++k) s += A[i * DIM + k] * inv[j][k];
            M3[i * DIM + j] = -s;
            M2[i * DIM + j] = -sqrtL * s;
        }
}

// ---------------------------------------------------------------------------
// Kernel 2: fully fused main loop. One block handles 16 batch rows across all
// 201 time steps (descending), using fp32 WMMA (16x16x4) for the MLP.
// ---------------------------------------------------------------------------
__global__ __launch_bounds__(BLOCK) void soc_main(
    const float* __restrict__ states,     // [201,4096,20]
    const float* __restrict__ noises,     // [200,4096,20]
    const float* __restrict__ controls,   // [200,4096,20]
    const float* __restrict__ log_weight, // [4096]
    const float* __restrict__ ts,         // [201]
    const float* __restrict__ sigma,      // [20,20]
    const float* __restrict__ P,          // [20,20]
    const float* __restrict__ Q,          // [20,20]
    const float* __restrict__ W1,         // [21,128]
    const float* __restrict__ b1,         // [128]
    const float* __restrict__ W2,         // [128,20]
    const float* __restrict__ b2,         // [20]
    const float* __restrict__ M2g,        // [20,20] precomputed
    const float* __restrict__ M3g,        // [20,20] precomputed
    float* __restrict__ partials)         // [256] per-block partial sums
{
    __shared__ float W1_s[24 * 128];   // K padded 21->24, zero-filled
    __shared__ float W2_s[128 * 32];   // N padded 20->32, zero-filled (unconditional B reads)
    __shared__ float P_s[400], M2_s[400], M3_s[400], Q_s[400], ST_s[400];
    __shared__ float b1_s[128], b2_s[20];
    __shared__ float ts_s[TN];
    __shared__ float weight_s[BTILE];
    __shared__ float tx_s[BTILE * 24];   // [16][24] (t, x0..x19, pad)
    __shared__ float big_s[4096];        // h [16][128] | GEMM2 partials [2][8][16][16] | final obj staging
    __shared__ float nV_s[BTILE * DIM];
    __shared__ float term_s[BTILE * DIM];
    __shared__ float acc_s[BTILE * DIM];
    __shared__ float diff_s[BTILE * DIM];
    __shared__ __align__(16) float xprev_s[BTILE * DIM];
    __shared__ __align__(16) float nprev_s[BTILE * DIM];
    __shared__ __align__(16) float cprev_s[BTILE * DIM];

    const int tid = threadIdx.x;
    const int w = tid >> 5;          // wave id 0..7
    const int lane = tid & 31;
    const int lhalf = lane >> 4;     // 0 = lanes 0-15, 1 = lanes 16-31
    const int lm = lane & 15;
    const int b0 = blockIdx.x * BTILE;

    // ---- one-time block init ----
    for (int idx = tid; idx < 24 * 128; idx += BLOCK) {
        int k = idx >> 7, n = idx & 127;
        W1_s[idx] = (k < 21) ? W1[k * 128 + n] : 0.0f;
    }
    for (int idx = tid; idx < 128 * 32; idx += BLOCK) {
        int k = idx >> 5, n = idx & 31;
        W2_s[idx] = (n < DIM) ? W2[k * DIM + n] : 0.0f;
    }
    for (int idx = tid; idx < 400; idx += BLOCK) {
        P_s[idx] = P[idx];
        Q_s[idx] = Q[idx];
        M2_s[idx] = M2g[idx];
        M3_s[idx] = M3g[idx];
        int i = idx / DIM, j = idx % DIM;
        ST_s[idx] = sigma[j * DIM + i];   // sigma.T
    }
    for (int idx = tid; idx < 128; idx += BLOCK) b1_s[idx] = b1[idx];
    if (tid < DIM) b2_s[tid] = b2[tid];
    for (int idx = tid; idx < TN; idx += BLOCK) ts_s[idx] = ts[idx];
    if (tid < BTILE) {
        weight_s[tid] = expf(log_weight[b0 + tid]);
        tx_s[tid * 24 + 21] = 0.0f;
        tx_s[tid * 24 + 22] = 0.0f;
        tx_s[tid * 24 + 23] = 0.0f;
    }
    // states[-1] tile (vectorized: 320 float4 = contiguous 1280 floats)
    {
        const float4* src = (const float4*)(states + ((size_t)NSTEPS * BATCH + b0) * DIM);
        for (int e = tid; e < 320; e += BLOCK) ((float4*)xprev_s)[e] = src[e];
        for (int e = tid; e < BTILE * DIM; e += BLOCK) acc_s[e] = 0.0f;
    }
    __syncthreads();
    // terminal = Q @ states[-1]
    for (int e = tid; e < BTILE * DIM; e += BLOCK) {
        int b = e / DIM, i = e % DIM;
        float s = 0.0f;
        for (int j = 0; j < DIM; ++j) s += Q_s[i * DIM + j] * xprev_s[b * DIM + j];
        term_s[e] = s;
    }

    float obj_acc = 0.0f;

    // ---- descending time loop: running suffix sum == reverse cumsum ----
    for (int t = NSTEPS; t >= 0; --t) {
        __syncthreads();
        // load tiles (float4 vectorized; each tile slab is contiguous & 16B aligned)
        {
            const float4* xs4 = (const float4*)(states + ((size_t)t * BATCH + b0) * DIM);
            const size_t poff = (t >= 1) ? ((size_t)(t - 1) * BATCH + b0) * DIM : 0;
            const float4* xp4 = (const float4*)(states + poff);
            const float4* np4 = (const float4*)(noises + poff);
            const float4* cp4 = (const float4*)(controls + poff);
            for (int e = tid; e < 320; e += BLOCK) {
                float4 xv = xs4[e];
                int b = e / 5, j0 = 4 * (e % 5);   // DIM=20: float4 never crosses a row
                float* dst = &tx_s[b * 24 + 1 + j0];
                dst[0] = xv.x; dst[1] = xv.y; dst[2] = xv.z; dst[3] = xv.w;
                if (t >= 1) {
                    ((float4*)xprev_s)[e] = xp4[e];
                    ((float4*)nprev_s)[e] = np4[e];
                    ((float4*)cprev_s)[e] = cp4[e];
                }
            }
        }
        if (tid < BTILE) tx_s[tid * 24 + 0] = ts_s[t];
        __syncthreads();

        // ---- GEMM1: h = tanh(tx[16x24] @ W1[24x128] + b1); wave w owns cols [16w,16w+16)
        v8f c1 = {};
        #pragma unroll
        for (int ks = 0; ks < 6; ++ks) {
            int k0 = 4 * ks + 2 * lhalf;
            v2f a, b;
            a.x = tx_s[lm * 24 + k0];
            a.y = tx_s[lm * 24 + k0 + 1];
            int col = 16 * w + lm;
            b.x = W1_s[k0 * 128 + col];
            b.y = W1_s[(k0 + 1) * 128 + col];
            c1 = __builtin_amdgcn_wmma_f32_16x16x4_f32(false, a, false, b,
                                                       (short)0, c1, false, false);
        }
        {
            int col = 16 * w + lm;
            float bb = b1_s[col];
            int mbase = 8 * lhalf;
            #pragma unroll
            for (int r = 0; r < 8; ++r)
                big_s[(mbase + r) * 128 + col] = tanh_fast(c1[r] + bb);
        }
        __syncthreads();

        // ---- GEMM2: nV = h[16x128] @ W2[128x32(pad)]; wave w owns K-chunk [16w,16w+16)
        v8f c2a = {}, c2b = {};
        #pragma unroll
        for (int ks = 0; ks < 4; ++ks) {
            int k0 = 16 * w + 4 * ks + 2 * lhalf;
            v2f a;
            a.x = big_s[lm * 128 + k0];
            a.y = big_s[lm * 128 + k0 + 1];
            v2f b;
            int col0 = lm;          // n-tile 0: cols 0..15
            b.x = W2_s[k0 * 32 + col0];
            b.y = W2_s[(k0 + 1) * 32 + col0];
            c2a = __builtin_amdgcn_wmma_f32_16x16x4_f32(false, a, false, b,
                                                        (short)0, c2a, false, false);
            int col1 = 16 + lm;     // n-tile 1: cols 16..31 (>=20 are zero-padded)
            b.x = W2_s[k0 * 32 + col1];
            b.y = W2_s[(k0 + 1) * 32 + col1];
            c2b = __builtin_amdgcn_wmma_f32_16x16x4_f32(false, a, false, b,
                                                        (short)0, c2b, false, false);
        }
        __syncthreads();  // all waves done reading h before partials overwrite big_s
        {
            int n = lm;
            int mbase = 8 * lhalf;
            #pragma unroll
            for (int r = 0; r < 8; ++r) {
                big_s[0 * 2048 + w * 256 + (mbase + r) * 16 + n] = c2a[r];
                big_s[1 * 2048 + w * 256 + (mbase + r) * 16 + n] = c2b[r];
            }
        }
        __syncthreads();
        // reduce 8 wave-partials -> nabla_V (+ b2)
        {
            int m = tid >> 4, n = tid & 15;
            #pragma unroll
            for (int nt = 0; nt < 2; ++nt) {
                int col = 16 * nt + n;
                if (col < DIM) {
                    float s = b2_s[col];
                    #pragma unroll
                    for (int ww = 0; ww < 8; ++ww)
                        s += big_s[nt * 2048 + ww * 256 + tid];
                    nV_s[m * DIM + col] = s;
                }
            }
        }
        __syncthreads();

        // ---- elementwise: diff = nV - (acc + terminal); acc += s[t] ----
        float dt = 0.0f, sdt = 0.0f;
        if (t >= 1) { dt = ts_s[t] - ts_s[t - 1]; sdt = sqrtf(dt); }
        for (int e = tid; e < BTILE * DIM; e += BLOCK) {
            int b = e / DIM, i = e % DIM;
            float accv = acc_s[e];
            diff_s[e] = nV_s[e] - (accv + term_s[e]);
            if (t >= 1) {
                float s1 = 0.0f, s2 = 0.0f, s3 = 0.0f;
                for (int j = 0; j < DIM; ++j) {
                    s1 += P_s[i * DIM + j]  * xprev_s[b * DIM + j];
                    s2 += M2_s[i * DIM + j] * nprev_s[b * DIM + j];
                    s3 += M3_s[i * DIM + j] * cprev_s[b * DIM + j];
                }
                acc_s[e] = accv + dt * (s1 + s3) + sdt * s2;
            }
        }
        __syncthreads();
        // y = sigma.T @ diff ; obj += y^2 * weight
        for (int e = tid; e < BTILE * DIM; e += BLOCK) {
            int b = e / DIM, i = e % DIM;
            float y = 0.0f;
            for (int j = 0; j < DIM; ++j) y += ST_s[i * DIM + j] * diff_s[b * DIM + j];
            obj_acc += y * y * weight_s[b];
        }
    }

    // deterministic fixed-order block reduction (big_s is dead here; reuse it)
    __syncthreads();
    big_s[tid] = obj_acc;
    __syncthreads();
    if (tid == 0) {
        float s = 0.0f;
        for (int i = 0; i < BLOCK; ++i) s += big_s[i];
        partials[blockIdx.x] = s;
    }
}

// ---------------------------------------------------------------------------
// Kernel 3: deterministic final reduction
// ---------------------------------------------------------------------------
__global__ void soc_finalize(const float* __restrict__ partials,
                             float* __restrict__ out) {
    if (threadIdx.x == 0 && blockIdx.x == 0) {
        float s = 0.0f;
        for (int i = 0; i < BATCH / BTILE; ++i) s += partials[i];
        out[0] = s / ((float)TN * (float)BATCH);
    }
}

extern "C" void kernel_launch(void* const* d_in, const int* in_sizes, int n_in,
                              void* d_out, int out_size, void* d_ws, size_t ws_size,
                              hipStream_t stream) {
    const float* states     = (const float*)d_in[0];
    const float* noises     = (const float*)d_in[1];
    const float* controls   = (const float*)d_in[2];
    const float* log_weight = (const float*)d_in[3];
    const float* ts         = (const float*)d_in[4];
    const float* sigma      = (const float*)d_in[5];
    const float* P          = (const float*)d_in[6];
    const float* A          = (const float*)d_in[7];
    const float* Q          = (const float*)d_in[8];
    const float* W1         = (const float*)d_in[9];
    const float* b1         = (const float*)d_in[10];
    const float* W2         = (const float*)d_in[11];
    const float* b2         = (const float*)d_in[12];

    float* ws = (float*)d_ws;
    float* M2 = ws;             // 400 floats
    float* M3 = ws + 400;       // 400 floats
    float* partials = ws + 800; // 256 floats

    soc_precompute<<<1, 32, 0, stream>>>(sigma, A, M2, M3);
    soc_main<<<BATCH / BTILE, BLOCK, 0, stream>>>(
        states, noises, controls, log_weight, ts, sigma, P, Q,
        W1, b1, W2, b2, M2, M3, partials);
    soc_finalize<<<1, 32, 0, stream>>>(partials, (float*)d_out);
}